// WritingLayerReLU_13563506721324
// MI455X (gfx1250) — compile-verified
//
#include <hip/hip_runtime.h>
#include <stdint.h>

// WritingLayerReLU for MI455X (gfx1250):
//   GEMM1: i = X[16384,1024] @ W^T[1024,2048], ReLU -> keys/vals
//   GEMM2: mem[b] = keys[b]^T @ vals[b]   (the scan is a batched GEMM)
// fp32 WMMA (V_WMMA_F32_16X16X4_F32) for exact fp32 accumulation.
// Global->LDS staging uses the CDNA5 async data mover
// (GLOBAL_LOAD_ASYNC_TO_LDS_B128, ASYNCcnt) with double-buffered LDS so the
// next chunk's copy runs underneath the current chunk's 64 WMMAs.

typedef __attribute__((ext_vector_type(2))) float v2f;
typedef __attribute__((ext_vector_type(8))) float v8f;

namespace {
constexpr int kB  = 16;
constexpr int kS  = 1024;
constexpr int kH  = 1024;
constexpr int kK1 = 1024;          // IN
constexpr int kM1 = kB * kS;       // 16384
constexpr int kN1 = 2 * kH;        // 2048
constexpr int BK  = 16;            // K-chunk staged in LDS
constexpr int LDP = 20;            // gemm1 LDS row stride (floats): 80B rows ->
                                   // 16B aligned, 20*r mod 64 banks all distinct
constexpr int P2  = 144;           // gemm2 LDS row stride (floats): 576B rows ->
                                   // 16B aligned; 144 mod 64 = 16 -> the four
                                   // 16-lane read groups hit disjoint bank quadrants
}

// CDNA5 async global->LDS copy, 16B per lane. LDS dest = low 32 bits of the
// generic shared-memory pointer (ISA 10.2: LDS aperture -> addr[31:0]).
__device__ __forceinline__ void async_ld_b128(uint32_t lds_off, const float* gaddr) {
    asm volatile("global_load_async_to_lds_b128 %0, %1, off"
                 :: "v"(lds_off), "v"(gaddr)
                 : "memory");
}
__device__ __forceinline__ void wait_async0() {
    asm volatile("s_wait_asynccnt 0x0" ::: "memory");
}
__device__ __forceinline__ uint32_t lds_off32(const void* p) {
    return (uint32_t)(uintptr_t)p;
}

// ---------------------------------------------------------------------------
// GEMM1: C[m,n] = sum_k X[m,k]*W[n,k]; ReLU; n<H -> keys else vals.
// 128 threads (4 waves), 128x128 block tile, 64x64 per wave (4x4 accums).
// LDS tiles [row][k], double buffered, filled by async b128 copies.
// ---------------------------------------------------------------------------
__global__ __launch_bounds__(128)
void wl_gemm1_relu(const float* __restrict__ X, const float* __restrict__ W,
                   float* __restrict__ keys, float* __restrict__ vals)
{
    __shared__ float As[2][128 * LDP];   // As[buf][m][k],  A[m,k]=X[m,k]
    __shared__ float Bs[2][128 * LDP];   // Bs[buf][n][k],  B[k,n]=W[n,k]

    const int tid  = threadIdx.x;
    const int wave = tid >> 5;
    const int lane = tid & 31;
    const int lr   = lane & 15;
    const int hi   = lane >> 4;

    const int mBlock = blockIdx.y * 128;
    const int nBlock = blockIdx.x * 128;
    const int waveM  = (wave >> 1) * 64;
    const int waveN  = (wave & 1) * 64;

    v8f acc[4][4];
#pragma unroll
    for (int i = 0; i < 4; ++i)
#pragma unroll
        for (int j = 0; j < 4; ++j)
#pragma unroll
            for (int e = 0; e < 8; ++e) acc[i][j][e] = 0.0f;

    // Thread tid owns LDS row tid of each tile: 16 floats = 4 async b128.
    const float* Xrow = X + (size_t)(mBlock + tid) * kK1;
    const float* Wrow = W + (size_t)(nBlock + tid) * kK1;
    const uint32_t aOff[2] = { lds_off32(&As[0][tid * LDP]),
                               lds_off32(&As[1][tid * LDP]) };
    const uint32_t bOff[2] = { lds_off32(&Bs[0][tid * LDP]),
                               lds_off32(&Bs[1][tid * LDP]) };

    auto issue = [&](int kc, int buf) {
#pragma unroll
        for (int q = 0; q < 4; ++q) {
            async_ld_b128(aOff[buf] + 16u * q, Xrow + kc + 4 * q);
            async_ld_b128(bOff[buf] + 16u * q, Wrow + kc + 4 * q);
        }
    };

    issue(0, 0);
    int cur = 0;
    for (int kc = 0; kc < kK1; kc += BK) {
        wait_async0();        // my async copies into buf[cur] landed
        __syncthreads();      // everyone's copies landed; prev reads of buf[cur^1] done
        if (kc + BK < kK1) issue(kc + BK, cur ^ 1);

        const float* Ab = As[cur];
        const float* Bb = Bs[cur];
#pragma unroll
        for (int kk = 0; kk < BK; kk += 4) {
            // fp32 16x16x4 fragment: lane<16 holds K=kk,kk+1; lane>=16 K=kk+2,kk+3
            const int k0 = kk + 2 * hi;
            v2f a[4], b[4];
#pragma unroll
            for (int i = 0; i < 4; ++i)
                a[i] = *(const v2f*)&Ab[(waveM + i * 16 + lr) * LDP + k0];
#pragma unroll
            for (int j = 0; j < 4; ++j)
                b[j] = *(const v2f*)&Bb[(waveN + j * 16 + lr) * LDP + k0];
#pragma unroll
            for (int i = 0; i < 4; ++i)
#pragma unroll
                for (int j = 0; j < 4; ++j)
                    acc[i][j] = __builtin_amdgcn_wmma_f32_16x16x4_f32(
                        false, a[i], false, b[j], (short)0, acc[i][j],
                        false, false);
        }
        cur ^= 1;
    }

    // n-tiles are 128 wide, H=1024 -> each block is entirely keys or vals.
    float* dst   = (nBlock < kH) ? keys : vals;
    const int n0 = (nBlock < kH) ? nBlock : (nBlock - kH);
#pragma unroll
    for (int i = 0; i < 4; ++i) {
        const int row0 = mBlock + waveM + i * 16 + 8 * hi;  // C: VGPR r -> M=r+8*hi
#pragma unroll
        for (int r = 0; r < 8; ++r) {
#pragma unroll
            for (int j = 0; j < 4; ++j) {
                const int col = n0 + waveN + j * 16 + lr;   // C: N = lane%16
                float v = acc[i][j][r];
                dst[(size_t)(row0 + r) * kH + col] = v > 0.0f ? v : 0.0f;
            }
        }
    }
}

// ---------------------------------------------------------------------------
// GEMM2: mem[b][i][j] = sum_t keys[b,t,i]*vals[b,t,j]  (K^T*V per batch).
// Chunk stored UNtransposed in LDS ([t][col], stride P2) so async b128 copies
// work; fragments read column-wise (two strided b32 -> ds_load_2addr_b32).
// ---------------------------------------------------------------------------
__global__ __launch_bounds__(128)
void wl_gemm2_mem(const float* __restrict__ keys, const float* __restrict__ vals,
                  float* __restrict__ mem)
{
    __shared__ float Ks[2][BK * P2];   // Ks[buf][t][i_local]
    __shared__ float Vs[2][BK * P2];   // Vs[buf][t][j_local]

    const int tid  = threadIdx.x;
    const int wave = tid >> 5;
    const int lane = tid & 31;
    const int lr   = lane & 15;
    const int hi   = lane >> 4;

    const int batch  = blockIdx.z;
    const int iBlock = blockIdx.y * 128;
    const int jBlock = blockIdx.x * 128;
    const int waveM  = (wave >> 1) * 64;
    const int waveN  = (wave & 1) * 64;

    const float* Kb = keys + (size_t)batch * kS * kH;
    const float* Vb = vals + (size_t)batch * kS * kH;

    v8f acc[4][4];
#pragma unroll
    for (int i = 0; i < 4; ++i)
#pragma unroll
        for (int j = 0; j < 4; ++j)
#pragma unroll
            for (int e = 0; e < 8; ++e) acc[i][j][e] = 0.0f;

    // 16 t-rows x 128 cols per chunk; thread -> (t row, 16-float col segment).
    const int trow = tid >> 3;        // 0..15
    const int cseg = (tid & 7) * 16;  // 0..112

    const float* kSrc0 = Kb + iBlock + cseg;   // + (kc+trow)*kH
    const float* vSrc0 = Vb + jBlock + cseg;
    const uint32_t kOff[2] = { lds_off32(&Ks[0][trow * P2 + cseg]),
                               lds_off32(&Ks[1][trow * P2 + cseg]) };
    const uint32_t vOff[2] = { lds_off32(&Vs[0][trow * P2 + cseg]),
                               lds_off32(&Vs[1][trow * P2 + cseg]) };

    auto issue = [&](int kc, int buf) {
        const size_t r = (size_t)(kc + trow) * kH;
#pragma unroll
        for (int q = 0; q < 4; ++q) {
            async_ld_b128(kOff[buf] + 16u * q, kSrc0 + r + 4 * q);
            async_ld_b128(vOff[buf] + 16u * q, vSrc0 + r + 4 * q);
        }
    };

    issue(0, 0);
    int cur = 0;
    for (int kc = 0; kc < kS; kc += BK) {
        wait_async0();
        __syncthreads();
        if (kc + BK < kS) issue(kc + BK, cur ^ 1);

        const float* Kt = Ks[cur];
        const float* Vt = Vs[cur];
#pragma unroll
        for (int kk = 0; kk < BK; kk += 4) {
            const int k0 = kk + 2 * hi;
            v2f a[4], b[4];
#pragma unroll
            for (int i = 0; i < 4; ++i) {          // A[m=i_local, k=t]
                const int c = waveM + i * 16 + lr;
                a[i][0] = Kt[k0 * P2 + c];
                a[i][1] = Kt[(k0 + 1) * P2 + c];
            }
#pragma unroll
            for (int j = 0; j < 4; ++j) {          // B[k=t, n=j_local]
                const int c = waveN + j * 16 + lr;
                b[j][0] = Vt[k0 * P2 + c];
                b[j][1] = Vt[(k0 + 1) * P2 + c];
            }
#pragma unroll
            for (int i = 0; i < 4; ++i)
#pragma unroll
                for (int j = 0; j < 4; ++j)
                    acc[i][j] = __builtin_amdgcn_wmma_f32_16x16x4_f32(
                        false, a[i], false, b[j], (short)0, acc[i][j],
                        false, false);
        }
        cur ^= 1;
    }

    float* out = mem + (size_t)batch * kH * kH;
#pragma unroll
    for (int i = 0; i < 4; ++i) {
        const int row0 = iBlock + waveM + i * 16 + 8 * hi;
#pragma unroll
        for (int r = 0; r < 8; ++r) {
#pragma unroll
            for (int j = 0; j < 4; ++j) {
                const int col = jBlock + waveN + j * 16 + lr;
                out[(size_t)(row0 + r) * kH + col] = acc[i][j][r];
            }
        }
    }
}

extern "C" void kernel_launch(void* const* d_in, const int* in_sizes, int n_in,
                              void* d_out, int out_size, void* d_ws, size_t ws_size,
                              hipStream_t stream) {
    (void)in_sizes; (void)n_in; (void)out_size; (void)d_ws; (void)ws_size;

    const float* X = (const float*)d_in[0];   // [B,S,IN]
    const float* W = (const float*)d_in[1];   // [2H,IN]

    // d_out = mem [B,H,H] ++ keys [B,S,H] ++ vals [B,S,H], all fp32
    float* mem  = (float*)d_out;
    float* keys = mem + (size_t)kB * kH * kH;
    float* vals = keys + (size_t)kB * kS * kH;

    dim3 g1(kN1 / 128, kM1 / 128);            // (16, 128)
    wl_gemm1_relu<<<g1, 128, 0, stream>>>(X, W, keys, vals);

    dim3 g2(kH / 128, kH / 128, kB);          // (8, 8, 16)
    wl_gemm2_mem<<<g2, 128, 0, stream>>>(keys, vals, mem);
}